// ConvLayerWithStyleMod_55516747268339
// MI455X (gfx1250) — compile-verified
//
#include <hip/hip_runtime.h>
#include <math.h>

// ---------------- problem constants ----------------
#define B_     8
#define C_     128
#define OC_    64
#define SD_    512
#define H_     128
#define KW_    3
#define H2_    257      // transposed-conv output size
#define OUT_   256      // final output size after blur
#define PP_    129      // phase-plane pitch (max phase dim)
#define PPSZ_  (PP_*PP_) // 16641 floats per (b,phase,o) plane

#define SCALE_  (0.02946278254943948f)   // 1/sqrt(128*9)
#define WGAIN_  (0.04419417382415922f)   // 1/sqrt(512)

typedef __attribute__((ext_vector_type(2))) float v2f;
typedef __attribute__((ext_vector_type(8))) float v8f;

// ---------------- kernel 1: style modulation s[b][c] ----------------
__global__ void k_style(const float* __restrict__ style,
                        const float* __restrict__ modw,
                        const float* __restrict__ modb,
                        float* __restrict__ s_ws) {
    int b = blockIdx.x;          // 8
    int c = threadIdx.x;         // 128
    const float* st = style + b * SD_;
    const float* mw = modw + c * SD_;
    float acc = 0.f;
#pragma unroll 4
    for (int k = 0; k < SD_; ++k) acc += st[k] * mw[k];
    s_ws[b * C_ + c] = acc * WGAIN_ + modb[c];
}

// ---------------- kernel 2: modulate+demodulate weights ----------------
// Output layout: W2[b][tap][kpair][o] as float2 (K-pair packed), i.e. float index
//   (((b*9 + tap)*64 + c/2)*64 + o)*2 + (c&1)
__global__ void k_wmod(const float* __restrict__ weight,
                       const float* __restrict__ s_ws,
                       float* __restrict__ W2) {
    int o = blockIdx.x;          // 64
    int b = blockIdx.y;          // 8
    int c = threadIdx.x;         // 128 (= K)
    float sv = s_ws[b * C_ + c] * SCALE_;
    const float* wp = weight + ((size_t)o * C_ + c) * (KW_ * KW_);
    float wv[9];
    float ss = 0.f;
#pragma unroll
    for (int t = 0; t < 9; ++t) { wv[t] = wp[t] * sv; ss += wv[t] * wv[t]; }
    __shared__ float red[C_];
    red[c] = ss;
    __syncthreads();
    for (int off = 64; off > 0; off >>= 1) {
        if (c < off) red[c] += red[c + off];
        __syncthreads();
    }
    float dem = rsqrtf(red[0] + 1e-8f);
#pragma unroll
    for (int t = 0; t < 9; ++t)
        W2[((((size_t)b * 9 + t) * 64 + (c >> 1)) * OC_ + o) * 2 + (c & 1)] = wv[t] * dem;
}

// ---------------- kernel 3: phase-specialized transposed conv via f32 WMMA --------------
// Phase (PI,PJ): output pixel (2i'+PI, 2j'+PJ). Dense sub-conv with TYxTX taps.
// grid: x = j'-tiles of 32, y = i', z = b. block: 256 thr = 8 waves,
// wave tile = 16 out-ch x 16 pixels, K = 128 in-channels.
template<int PI, int PJ>
__global__ void __launch_bounds__(256) k_conv(const float* __restrict__ x,
                                              const float* __restrict__ W2,
                                              float* __restrict__ conv_ws) {
    constexpr int TY = PI ? 1 : 2;
    constexpr int TX = PJ ? 1 : 2;
    constexpr int NT = TY * TX;
    constexpr int WD = 32 + (PJ ? 0 : 1);   // LDS tile width
    constexpr int NJ = PJ ? 128 : 129;
    constexpr int PH = PI * 2 + PJ;

    const int b  = blockIdx.z;
    const int ip = blockIdx.y;              // i'
    const int j0 = blockIdx.x * 32;         // j' base
    const int r0 = ip - (PI ? 0 : 1);       // first x row needed
    const int c0 = j0 - (PJ ? 0 : 1);       // first x col needed

    // xs[kpair][trow][col] as float2 (x[2kp], x[2kp+1]) : 64*2*34*8B = 34 KB
    __shared__ v2f xs[64 * 2 * 34];

    const float* xb = x + (size_t)b * C_ * H_ * H_;
    const int total = 64 * TY * WD;
    for (int idx = threadIdx.x; idx < total; idx += 256) {
        int col = idx % WD;
        int rem = idx / WD;
        int t   = rem % TY;
        int kp  = rem / TY;
        int gr = r0 + t, gc = c0 + col;
        v2f v = {};
        if (gr >= 0 && gr < H_ && gc >= 0 && gc < H_) {
            const float* p = xb + ((size_t)(2 * kp) * H_ + gr) * H_ + gc;
            v.x = p[0];
            v.y = p[(size_t)H_ * H_];
        }
        xs[(kp * 2 + t) * 34 + col] = v;
    }
    __syncthreads();

    const int lane = threadIdx.x & 31;
    const int wid  = threadIdx.x >> 5;   // 0..7
    const int mW   = wid & 3;            // out-channel tile (4 x 16 = 64)
    const int nW   = wid >> 2;           // pixel tile (2 x 16 = 32)
    const int ln   = lane & 15;
    const int khp  = lane >> 4;          // K-pair half: lanes 0-15 -> K{0,1}, 16-31 -> K{2,3}
    const int m    = mW * 16 + ln;

    const v2f* Wb = (const v2f*)W2 + (size_t)b * 9 * 64 * OC_;

    v8f acc0 = {}, acc1 = {};
#pragma unroll 2
    for (int kc = 0; kc < C_; kc += 4) {
        const int kp = (kc >> 1) + khp;
        v2f a[NT], bf[NT];
#pragma unroll
        for (int t = 0; t < NT; ++t) {
            const int ty = t / TX, tx = t % TX;
            // LDS row ty=0 is x row i'-1 -> wmod tap qy=2; ty=TY-1 is x row i' -> qy=0
            const int ky  = PI ? 1 : (2 - 2 * ty);
            const int kx  = PJ ? 1 : (2 - 2 * tx);
            const int tap = ky * 3 + kx;
            a[t]  = Wb[((size_t)tap * 64 + kp) * OC_ + m];          // global_load_b64
            bf[t] = xs[(kp * 2 + ty) * 34 + (nW * 16 + ln + tx)];   // ds_load_b64
        }
#pragma unroll
        for (int t = 0; t < NT; ++t) {
            if (t & 1)
                acc1 = __builtin_amdgcn_wmma_f32_16x16x4_f32(
                         false, a[t], false, bf[t], (short)0, acc1, false, false);
            else
                acc0 = __builtin_amdgcn_wmma_f32_16x16x4_f32(
                         false, a[t], false, bf[t], (short)0, acc0, false, false);
        }
    }

    // D layout: lane l holds N = l&15; VGPR r -> M = r + 8*(l>=16)
    const int jp = j0 + nW * 16 + ln;
    if (jp < NJ) {
        const int mhi = khp * 8;
        float* op = conv_ws + (((size_t)(b * 4 + PH) * OC_ + mW * 16 + mhi) * PPSZ_)
                  + (size_t)ip * PP_ + jp;
#pragma unroll
        for (int r = 0; r < 8; ++r) {
            float v = (NT > 1) ? (acc0[r] + acc1[r]) : acc0[r];
            op[(size_t)r * PPSZ_] = v;
        }
    }
}

// ---------------- kernel 4: 4x4 FIR blur (pad 1), gathers from phase planes --------------
__global__ void k_blur(const float* __restrict__ conv_ws, float* __restrict__ out) {
    const int y = blockIdx.x;    // 256
    const int o = blockIdx.y;    // 64
    const int b = blockIdx.z;    // 8
    const int xg = threadIdx.x;  // 256
    float fy[4] = {1.f, 3.f, 3.f, 1.f};
    float acc = 0.f;
#pragma unroll
    for (int ky = 0; ky < 4; ++ky) {
        int u = y + ky - 1;
        if (u < 0 || u >= H2_) continue;
        int pi = u & 1, i2 = u >> 1;
#pragma unroll
        for (int kx = 0; kx < 4; ++kx) {
            int v = xg + kx - 1;
            if (v < 0 || v >= H2_) continue;
            int pj = v & 1, j2 = v >> 1;
            int ph = pi * 2 + pj;
            float cv = conv_ws[(((size_t)(b * 4 + ph) * OC_ + o) * PPSZ_)
                               + (size_t)i2 * PP_ + j2];
            acc += fy[ky] * fy[kx] * cv;
        }
    }
    out[(((size_t)b * OC_ + o) * OUT_ + y) * OUT_ + xg] = acc * (1.f / 16.f);
}

// ---------------- launch ----------------
extern "C" void kernel_launch(void* const* d_in, const int* in_sizes, int n_in,
                              void* d_out, int out_size, void* d_ws, size_t ws_size,
                              hipStream_t stream) {
    const float* x      = (const float*)d_in[0];
    const float* style  = (const float*)d_in[1];
    const float* weight = (const float*)d_in[2];
    const float* modw   = (const float*)d_in[3];
    const float* modb   = (const float*)d_in[4];
    float* out = (float*)d_out;
    float* ws  = (float*)d_ws;

    float* s_ws    = ws;                       // 8*128          = 1024 floats
    float* W2_ws   = ws + 1024;                // 8*9*64*64*2    = 589824 floats
    float* conv_ws = ws + 1024 + 589824;       // 32*64*129*129  = ~136 MB

    k_style<<<dim3(B_), dim3(C_), 0, stream>>>(style, modw, modb, s_ws);
    k_wmod<<<dim3(OC_, B_), dim3(C_), 0, stream>>>(weight, s_ws, W2_ws);

    // phase (PI,PJ): grid x = ceil(NJ/32), y = NI, z = b
    k_conv<0,0><<<dim3(5, 129, B_), dim3(256), 0, stream>>>(x, W2_ws, conv_ws);
    k_conv<0,1><<<dim3(4, 129, B_), dim3(256), 0, stream>>>(x, W2_ws, conv_ws);
    k_conv<1,0><<<dim3(5, 128, B_), dim3(256), 0, stream>>>(x, W2_ws, conv_ws);
    k_conv<1,1><<<dim3(4, 128, B_), dim3(256), 0, stream>>>(x, W2_ws, conv_ws);

    k_blur<<<dim3(OUT_, OC_, B_), dim3(OUT_), 0, stream>>>(conv_ws, out);
}